// PretrainedModel_35691178230073
// MI455X (gfx1250) — compile-verified
//
#include <hip/hip_runtime.h>

// CDNA5 / gfx1250, wave32. Layer-2 GEMM on V_WMMA_F32_16X16X4_F32.

typedef __attribute__((ext_vector_type(2))) float v2f;
typedef __attribute__((ext_vector_type(8))) float v8f;

#define HID 64
#define SAMPLES 512
#define WAVES 8
#define W2T_STRIDE 66   // padded LDS stride (floats), keeps 8B alignment, kills bank conflicts

__global__ __launch_bounds__(256) void mlp_task_mse_kernel(
    const float* __restrict__ labels,   // [T, 512, 1]
    const float* __restrict__ inputs,   // [T, 512, 1]
    const float* __restrict__ W1,       // [1, 64]
    const float* __restrict__ b1,       // [64]
    const float* __restrict__ W2,       // [64, 64] row-major [k][n]
    const float* __restrict__ b2,       // [64]
    const float* __restrict__ W3,       // [64, 1]
    const float* __restrict__ b3,       // [1]
    float* __restrict__ out)            // [T]
{
    __shared__ float sW1[HID];
    __shared__ float sB1[HID];
    __shared__ float sB2[HID];
    __shared__ float sW3[HID];
    __shared__ float sW2T[HID * W2T_STRIDE];  // [n][k] transposed + padded
    __shared__ float sPartial[WAVES];

    const int tid  = threadIdx.x;
    const int task = blockIdx.x;

    if (tid < HID) {
        sW1[tid] = W1[tid];
        sB1[tid] = b1[tid];
        sB2[tid] = b2[tid];
        sW3[tid] = W3[tid];
    }
    // W2[k][n] -> sW2T[n][k] (coalesced global read, scattered LDS write)
    for (int i = tid; i < HID * HID; i += 256) {
        const int k = i >> 6;
        const int n = i & 63;
        sW2T[n * W2T_STRIDE + k] = W2[i];
    }
    __syncthreads();

    const int wave = tid >> 5;
    const int lane = tid & 31;
    const int half = lane >> 4;      // H: 0 = lanes 0-15, 1 = lanes 16-31
    const int ln16 = lane & 15;
    const float b3v = b3[0];

    const long base = (long)task * SAMPLES + (long)wave * (SAMPLES / WAVES);

    float waveSum = 0.0f;

    for (int tile = 0; tile < 4; ++tile) {
        const long m0 = base + tile * 16;

        // x for this lane's sample (lanes L and L+16 read the same sample)
        const float xm = inputs[m0 + ln16];

        // ---- Layer 1 directly into WMMA f32 16x16x4 A-layout ----
        // A VGPR pair for k-step s: lane holds h1[m, 4s+2H] , h1[m, 4s+2H+1]
        v2f a[16];
#pragma unroll
        for (int s = 0; s < 16; ++s) {
            const int k0 = 4 * s + 2 * half;                // even -> 8B aligned
            const v2f w = *(const v2f*)&sW1[k0];
            const v2f c = *(const v2f*)&sB1[k0];
            a[s].x = fmaxf(fmaf(xm, w.x, c.x), 0.0f);
            a[s].y = fmaxf(fmaf(xm, w.y, c.y), 0.0f);
        }

        // ---- Layer 2: h2_pre = h1 @ W2 via 64 WMMAs (4 indep acc chains) ----
        v8f acc[4];
#pragma unroll
        for (int t = 0; t < 4; ++t)
#pragma unroll
            for (int r = 0; r < 8; ++r)
                acc[t][r] = 0.0f;

#pragma unroll
        for (int s = 0; s < 16; ++s) {
            const int row = 4 * s + 2 * half;               // B rows for this lane half
#pragma unroll
            for (int t = 0; t < 4; ++t) {
                const int n = t * 16 + ln16;                // output column
                const v2f b = *(const v2f*)&sW2T[n * W2T_STRIDE + row]; // ds_load_b64
                acc[t] = __builtin_amdgcn_wmma_f32_16x16x4_f32(
                    /*neg_a=*/false, a[s],
                    /*neg_b=*/false, b,
                    /*c_mod=*/(short)0, acc[t],
                    /*reuse_a=*/false, /*reuse_b=*/false);
            }
        }

        // ---- Layer 2 bias+ReLU and Layer 3 partial dot, in C/D layout ----
        // acc[t][r] at this lane = h2_pre[sample M = r + 8*half, j = 16t + ln16]
        float p[8];
#pragma unroll
        for (int r = 0; r < 8; ++r) p[r] = 0.0f;
#pragma unroll
        for (int t = 0; t < 4; ++t) {
            const int j = t * 16 + ln16;
            const float bj = sB2[j];
            const float wj = sW3[j];
#pragma unroll
            for (int r = 0; r < 8; ++r) {
                const float h2 = fmaxf(acc[t][r] + bj, 0.0f);
                p[r] = fmaf(h2, wj, p[r]);
            }
        }

        // Reduce the layer-3 dot over the 16 lanes of each half (masks < 16
        // keep the two halves independent in wave32).
#pragma unroll
        for (int r = 0; r < 8; ++r) {
            float v = p[r];
            v += __shfl_xor(v, 1, 32);
            v += __shfl_xor(v, 2, 32);
            v += __shfl_xor(v, 4, 32);
            v += __shfl_xor(v, 8, 32);
            p[r] = v;
        }

        // Select this lane's sample prediction: lower half holds samples 0-7,
        // upper half samples 8-15; lane picks p[lane&7] via cndmask ladder.
        float pred = p[0];
#pragma unroll
        for (int r = 1; r < 8; ++r)
            pred = ((lane & 7) == r) ? p[r] : pred;
        pred += b3v;

        const bool active    = (lane & 15) < 8;         // 16 active lanes = 16 samples
        const int  sampleIdx = (lane & 7) + 8 * half;
        const float lab = labels[m0 + sampleIdx];       // always in-range
        const float err = active ? (lab - pred) : 0.0f;
        float sq = err * err;

        // full-wave sum of the 16 squared errors
        sq += __shfl_xor(sq, 1, 32);
        sq += __shfl_xor(sq, 2, 32);
        sq += __shfl_xor(sq, 4, 32);
        sq += __shfl_xor(sq, 8, 32);
        sq += __shfl_xor(sq, 16, 32);
        waveSum += sq;
    }

    if (lane == 0) sPartial[wave] = waveSum;
    __syncthreads();
    if (tid == 0) {
        float total = 0.0f;
#pragma unroll
        for (int w = 0; w < WAVES; ++w) total += sPartial[w];
        out[task] = total * (1.0f / (float)SAMPLES);
    }
}

extern "C" void kernel_launch(void* const* d_in, const int* in_sizes, int n_in,
                              void* d_out, int out_size, void* d_ws, size_t ws_size,
                              hipStream_t stream) {
    const float* labels = (const float*)d_in[0];
    const float* inputs = (const float*)d_in[1];
    const float* W1     = (const float*)d_in[2];
    const float* b1     = (const float*)d_in[3];
    const float* W2     = (const float*)d_in[4];
    const float* b2     = (const float*)d_in[5];
    const float* W3     = (const float*)d_in[6];
    const float* b3     = (const float*)d_in[7];
    float* out = (float*)d_out;

    const int n_tasks = out_size;  // 4096 in the reference
    mlp_task_mse_kernel<<<n_tasks, 256, 0, stream>>>(
        labels, inputs, W1, b1, W2, b2, W3, b3, out);
}